// HeteroGNN_59261958750802
// MI455X (gfx1250) — compile-verified
//
#include <hip/hip_runtime.h>
#include <math.h>

#define N_USER 50000
#define N_BOOK 25000
#define NEDGE  400000
#define HID    64
#define EDIM   32
#define NOUT   8

typedef __attribute__((ext_vector_type(2))) float v2f;
typedef __attribute__((ext_vector_type(8))) float v8f;

// ---------------------------------------------------------------------------
// out[N,64] = X[N,K] @ W[64,K]^T (+ bias), fp32 WMMA 16x16x4.
// One wave computes a full 16x64 output stripe (4 column tiles), so the A
// operand (X rows) is loaded once per k-step and reused by 4 WMMAs.
// Tile index is wave-uniform so EXEC==all-1s at every WMMA.
// A: lanes0-15 K={k,k+1}, lanes16-31 K={k+2,k+3} (v2f per lane).
// B mirrors A with N=lane&15. C: vgpr r -> row r + 8*(lane>=16), col lane&15.
// ---------------------------------------------------------------------------
template <int K>
__global__ void gemm_wmma_f32(const float* __restrict__ X,
                              const float* __restrict__ W,
                              const float* __restrict__ bias,
                              float* __restrict__ out, int N) {
  const int lane = threadIdx.x & 31;
  const int wave = (blockIdx.x * blockDim.x + threadIdx.x) >> 5;
  const int rowTiles = (N + 15) >> 4;
  if (wave >= rowTiles) return;            // wave-uniform exit

  const int row0 = wave << 4;
  const int half = lane >> 4;              // 0: K pair {0,1}; 1: K pair {2,3}
  const int l15  = lane & 15;
  int arow = row0 + l15;
  if (arow > N - 1) arow = N - 1;          // clamp (stores are guarded)

  v8f c0 = {0.f,0.f,0.f,0.f,0.f,0.f,0.f,0.f};
  v8f c1 = c0, c2 = c0, c3 = c0;
  const float* wp0 = W + (size_t)(l15)      * K;   // col tile 0
  const float* wp1 = W + (size_t)(16 + l15) * K;   // col tile 1
  const float* wp2 = W + (size_t)(32 + l15) * K;   // col tile 2
  const float* wp3 = W + (size_t)(48 + l15) * K;   // col tile 3
  const float* xp  = X + (size_t)arow * K + half * 2;
#pragma unroll
  for (int k = 0; k < K; k += 4) {
    v2f a  = *(const v2f*)(xp + k);
    v2f b0 = *(const v2f*)(wp0 + k + half * 2);
    v2f b1 = *(const v2f*)(wp1 + k + half * 2);
    v2f b2 = *(const v2f*)(wp2 + k + half * 2);
    v2f b3 = *(const v2f*)(wp3 + k + half * 2);
    c0 = __builtin_amdgcn_wmma_f32_16x16x4_f32(false, a, false, b0, (short)0, c0, false, false);
    c1 = __builtin_amdgcn_wmma_f32_16x16x4_f32(false, a, false, b1, (short)0, c1, false, false);
    c2 = __builtin_amdgcn_wmma_f32_16x16x4_f32(false, a, false, b2, (short)0, c2, false, false);
    c3 = __builtin_amdgcn_wmma_f32_16x16x4_f32(false, a, false, b3, (short)0, c3, false, false);
  }
  float bv0 = bias ? bias[l15]      : 0.f;
  float bv1 = bias ? bias[16 + l15] : 0.f;
  float bv2 = bias ? bias[32 + l15] : 0.f;
  float bv3 = bias ? bias[48 + l15] : 0.f;
#pragma unroll
  for (int r = 0; r < 8; ++r) {
    int orow = row0 + r + half * 8;
    if (orow < N) {
      float* op = out + (size_t)orow * HID + l15;
      op[0]  = c0[r] + bv0;
      op[16] = c1[r] + bv1;
      op[32] = c2[r] + bv2;
      op[48] = c3[r] + bv3;
    }
  }
}

// ---------------------------------------------------------------------------
// Segment-softmax state init: amax=-inf, den=0, acc=0.
// ---------------------------------------------------------------------------
__global__ void init_seg_kernel(float* __restrict__ amax, float* __restrict__ den,
                                float* __restrict__ acc, int Nd) {
  int t = blockIdx.x * blockDim.x + threadIdx.x;
  if (t < Nd) { amax[t] = -__builtin_inff(); den[t] = 0.f; }
  if (t < Nd * HID) acc[t] = 0.f;
}

// ---------------------------------------------------------------------------
// Pass 1: alpha[e] = dot(q[dst], k[src]+eemb[e]) / sqrt(64); atomic max per dst.
// ---------------------------------------------------------------------------
__global__ void edge_alpha_kernel(const int* __restrict__ ei,
                                  const float* __restrict__ q,
                                  const float* __restrict__ k,
                                  const float* __restrict__ eemb,
                                  float* __restrict__ alpha,
                                  float* __restrict__ amax) {
  int e = blockIdx.x * blockDim.x + threadIdx.x;
  if (e >= NEDGE) return;
  int s = ei[e];
  int d = ei[NEDGE + e];
  const float4* qp = (const float4*)(q + (size_t)d * HID);
  const float4* kp = (const float4*)(k + (size_t)s * HID);
  const float4* ep = (const float4*)(eemb + (size_t)e * HID);
  float acc = 0.f;
#pragma unroll
  for (int c = 0; c < HID / 4; ++c) {
    float4 qv = qp[c], kv = kp[c], ev = ep[c];
    acc += qv.x * (kv.x + ev.x) + qv.y * (kv.y + ev.y) +
           qv.z * (kv.z + ev.z) + qv.w * (kv.w + ev.w);
  }
  float a = acc * 0.125f;  // 1/sqrt(HID)
  alpha[e] = a;
  // float atomic-max via signed/unsigned ordering trick (amax init = -inf)
  if (a >= 0.f) atomicMax((int*)(amax + d), __float_as_int(a));
  else          atomicMin((unsigned int*)(amax + d), __float_as_uint(a));
}

// ---------------------------------------------------------------------------
// Pass 2: ex = exp(alpha - amax[d]); den[d] += ex; acc[d,:] += (v[s,:]+e[:])*ex.
// 64 threads per edge -> coalesced channel access.
// ---------------------------------------------------------------------------
__global__ void edge_accum_kernel(const int* __restrict__ ei,
                                  const float* __restrict__ v,
                                  const float* __restrict__ eemb,
                                  const float* __restrict__ alpha,
                                  const float* __restrict__ amax,
                                  float* __restrict__ den,
                                  float* __restrict__ acc) {
  long long t = (long long)blockIdx.x * blockDim.x + threadIdx.x;
  int e = (int)(t >> 6);
  int c = (int)(t & 63);
  if (e >= NEDGE) return;
  int s = ei[e];
  int d = ei[NEDGE + e];
  float ex = __expf(alpha[e] - amax[d]);
  if (c == 0) atomicAdd(den + d, ex);
  float m = (v[(size_t)s * HID + c] + eemb[(size_t)e * HID + c]) * ex;
  atomicAdd(acc + (size_t)d * HID + c, m);
}

// ---------------------------------------------------------------------------
// out = relu(acc/den + skip); den==0 (no in-edges) -> message is 0.
// ---------------------------------------------------------------------------
__global__ void finalize_kernel(const float* __restrict__ acc,
                                const float* __restrict__ den,
                                const float* __restrict__ skip,
                                float* __restrict__ xout, int Nd) {
  int t = blockIdx.x * blockDim.x + threadIdx.x;
  if (t >= Nd * HID) return;
  int n = t >> 6;
  float ds = den[n];
  float m = ds > 0.f ? acc[t] / ds : 0.f;
  float r = m + skip[t];
  xout[t] = r > 0.f ? r : 0.f;
}

// ---------------------------------------------------------------------------
// Head: out[n,j] = sigmoid(dot(xb[n,:], W[j,:]) + b[j]),  j in [0,8)
// ---------------------------------------------------------------------------
__global__ void head_kernel(const float* __restrict__ xb,
                            const float* __restrict__ W,
                            const float* __restrict__ b,
                            float* __restrict__ out) {
  int t = blockIdx.x * blockDim.x + threadIdx.x;
  if (t >= N_BOOK * NOUT) return;
  int n = t >> 3, j = t & 7;
  const float4* xp = (const float4*)(xb + (size_t)n * HID);
  const float4* wp = (const float4*)(W + (size_t)j * HID);
  float acc = 0.f;
#pragma unroll
  for (int c = 0; c < HID / 4; ++c) {
    float4 x = xp[c], w = wp[c];
    acc += x.x * w.x + x.y * w.y + x.z * w.z + x.w * w.w;
  }
  acc += b[j];
  out[t] = 1.f / (1.f + __expf(-acc));
}

// ---------------------------------------------------------------------------
// Host-side orchestration
// ---------------------------------------------------------------------------
namespace {

struct ConvP {
  const float *qW, *qb, *kW, *kb, *vW, *vb, *eW, *sW, *sb;
};

struct Ws {
  float *q, *k, *v, *skip, *e, *alpha, *amax, *den, *acc;
};

inline dim3 gemm_grid(int N) {
  int tiles = (N + 15) / 16;            // one wave per 16-row stripe
  return dim3((tiles + 3) / 4);         // 4 waves (128 thr) per block
}

void run_conv(const float* xsrc, int Ns, const float* xdst, int Nd,
              const int* ei, const float* ea, const ConvP& p,
              float* xout, const Ws& w, hipStream_t stream) {
  const dim3 blk(128);
  gemm_wmma_f32<HID><<<gemm_grid(Nd), blk, 0, stream>>>(xdst, p.qW, p.qb, w.q, Nd);
  gemm_wmma_f32<HID><<<gemm_grid(Ns), blk, 0, stream>>>(xsrc, p.kW, p.kb, w.k, Ns);
  gemm_wmma_f32<HID><<<gemm_grid(Ns), blk, 0, stream>>>(xsrc, p.vW, p.vb, w.v, Ns);
  gemm_wmma_f32<HID><<<gemm_grid(Nd), blk, 0, stream>>>(xdst, p.sW, p.sb, w.skip, Nd);
  gemm_wmma_f32<EDIM><<<gemm_grid(NEDGE), blk, 0, stream>>>(ea, p.eW, nullptr, w.e, NEDGE);

  init_seg_kernel<<<(Nd * HID + 255) / 256, 256, 0, stream>>>(w.amax, w.den, w.acc, Nd);
  edge_alpha_kernel<<<(NEDGE + 255) / 256, 256, 0, stream>>>(ei, w.q, w.k, w.e, w.alpha, w.amax);
  edge_accum_kernel<<<((long long)NEDGE * HID + 255) / 256, 256, 0, stream>>>(
      ei, w.v, w.e, w.alpha, w.amax, w.den, w.acc);
  finalize_kernel<<<(Nd * HID + 255) / 256, 256, 0, stream>>>(w.acc, w.den, w.skip, xout, Nd);
}

}  // namespace

extern "C" void kernel_launch(void* const* d_in, const int* in_sizes, int n_in,
                              void* d_out, int out_size, void* d_ws, size_t ws_size,
                              hipStream_t stream) {
  const float* x_user = (const float*)d_in[0];
  const float* x_book = (const float*)d_in[1];
  const int*   ei_ub  = (const int*)d_in[2];
  const int*   ei_bu  = (const int*)d_in[3];
  const float* ea_ub  = (const float*)d_in[4];
  const float* ea_bu  = (const float*)d_in[5];

  // params flattened in dict-insertion order:
  // per conv (layer-major, ub then bu): qW,qb,kW,kb,vW,vb,eW,sW,sb  (9 arrays)
  auto conv_params = [&](int layer, int dir) {
    int base = 6 + (layer * 2 + dir) * 9;
    ConvP p;
    p.qW = (const float*)d_in[base + 0]; p.qb = (const float*)d_in[base + 1];
    p.kW = (const float*)d_in[base + 2]; p.kb = (const float*)d_in[base + 3];
    p.vW = (const float*)d_in[base + 4]; p.vb = (const float*)d_in[base + 5];
    p.eW = (const float*)d_in[base + 6];
    p.sW = (const float*)d_in[base + 7]; p.sb = (const float*)d_in[base + 8];
    return p;
  };
  const float* linW = (const float*)d_in[42];
  const float* linb = (const float*)d_in[43];

  // Workspace carve-up (fp32 elements)
  const size_t XU = (size_t)N_USER * HID;   // 3.2M
  const size_t XB = (size_t)N_BOOK * HID;   // 1.6M
  float* ws = (float*)d_ws;
  size_t off = 0;
  float* xuA = ws + off; off += XU;
  float* xuB = ws + off; off += XU;
  float* xbA = ws + off; off += XB;
  float* xbB = ws + off; off += XB;
  Ws w;
  w.q    = ws + off; off += XU;             // sized for max(N_USER, N_BOOK)
  w.k    = ws + off; off += XU;
  w.v    = ws + off; off += XU;
  w.skip = ws + off; off += XU;
  w.e    = ws + off; off += (size_t)NEDGE * HID;
  w.alpha= ws + off; off += (size_t)NEDGE;
  w.amax = ws + off; off += (size_t)N_USER;
  w.den  = ws + off; off += (size_t)N_USER;
  w.acc  = ws + off; off += XU;

  hipMemcpyAsync(xuA, x_user, XU * sizeof(float), hipMemcpyDeviceToDevice, stream);
  hipMemcpyAsync(xbA, x_book, XB * sizeof(float), hipMemcpyDeviceToDevice, stream);

  float *xu = xuA, *xb = xbA, *xuN = xuB, *xbN = xbB;
  for (int l = 0; l < 2; ++l) {
    // nb = relu(tconv(xu -> xb));  nu = relu(tconv(xb -> xu)); both use OLD xu/xb
    run_conv(xu, N_USER, xb, N_BOOK, ei_ub, ea_ub, conv_params(l, 0), xbN, w, stream);
    run_conv(xb, N_BOOK, xu, N_USER, ei_bu, ea_bu, conv_params(l, 1), xuN, w, stream);
    float* t;
    t = xu; xu = xuN; xuN = t;
    t = xb; xb = xbN; xbN = t;
  }

  head_kernel<<<(N_BOOK * NOUT + 255) / 256, 256, 0, stream>>>(xb, linW, linb, (float*)d_out);
}